// MambaLayer_33500744908982
// MI455X (gfx1250) — compile-verified
//
#include <hip/hip_runtime.h>
#include <math.h>

typedef float v2f __attribute__((ext_vector_type(2)));
typedef float v8f __attribute__((ext_vector_type(8)));

#define DIMC   128
#define DST    16
#define DIN    256
#define DTR    8
#define SEQL   2048
#define NB     2
#define NDIR   4
#define MROWS  (NDIR*NB*SEQL)   /* 16384 rows for mamba GEMMs */
#define MACT   (NB*SEQL)        /* 4096 rows for nin GEMMs    */

// ---------------------------------------------------------------------------
// f32 WMMA micro-op: D(16x16) += A(16x4) * B(4x16), fp32 throughout.
// A frag (per ISA 7.12.2): lane l -> row m = l&15, ks = (l>>4)*2; a = {A[m,k0+ks], A[m,k0+ks+1]}
// B frag (mirrored):       lane l -> col n = l&15, ks = (l>>4)*2; b = {B[k0+ks,n], B[k0+ks+1,n]}
// D frag: lane l -> col n = l&15, VGPR r -> row m = r + ((l>>4)<<3)
// ---------------------------------------------------------------------------
static __device__ __forceinline__ v8f wmma4(v8f acc, v2f a, v2f b) {
  return __builtin_amdgcn_wmma_f32_16x16x4_f32(false, a, false, b, (short)0, acc,
                                               false, false);
}

// ===========================================================================
// K1/K9: Y = silu(GRN(X @ W^T + bias))   (M=4096, N=K=128)
// xmode 0: X is the raw input laid out (SEQ,BATCH,DIM) -> row m=(b,n) gathers
//          X[(n*2+b)*128+c].  xmode 1: X row-major [m*128+c].
// block = 128 threads (4 waves); block computes a 16x128 stripe.
// ===========================================================================
__global__ void k_nin_grn_silu(const float* __restrict__ X, int xmode,
                               const float* __restrict__ W,
                               const float* __restrict__ bias,
                               const float* __restrict__ gamma,
                               const float* __restrict__ beta,
                               float* __restrict__ out)
{
  __shared__ float tile[16][128];
  __shared__ float gxs[16];
  const int tid  = threadIdx.x;
  const int lane = tid & 31;
  const int wv   = tid >> 5;          // 0..3
  const int half = lane >> 4;         // 0/1
  const int l15  = lane & 15;

  const int m = blockIdx.x * 16 + l15;            // A row for this lane
  const float* arow = (xmode == 0)
      ? X + ((size_t)(m & (SEQL-1)) * NB + (m >> 11)) * DIMC
      : X + (size_t)m * DIMC;

  #pragma unroll
  for (int sub = 0; sub < 2; ++sub) {
    const int n0 = wv * 32 + sub * 16;
    const int n  = n0 + l15;
    v8f acc = {};
    #pragma unroll 4
    for (int k0 = 0; k0 < DIMC; k0 += 4) {
      const int k = k0 + half * 2;
      v2f a;  a.x = arow[k];            a.y = arow[k + 1];
      v2f b;  b.x = W[n * DIMC + k];    b.y = W[n * DIMC + k + 1];
      acc = wmma4(acc, a, b);
    }
    const int mr = half << 3;
    const float bv = bias[n];
    #pragma unroll
    for (int r = 0; r < 8; ++r) tile[mr + r][n] = acc[r] + bv;
  }
  __syncthreads();

  // GRN + SiLU: 8 threads per row, 16 cols each
  const int r  = tid >> 3;
  const int cg = tid & 7;
  float p = 0.f;
  #pragma unroll
  for (int j = 0; j < 16; ++j) { float v = tile[r][cg * 16 + j]; p += v * v; }
  p += __shfl_xor(p, 4); p += __shfl_xor(p, 2); p += __shfl_xor(p, 1);
  if (cg == 0) gxs[r] = sqrtf(p);
  __syncthreads();

  const float gx  = gxs[r];
  const float nx  = gx / (gx + 1e-6f);
  const float inv = 1.0f / fmaxf(gx, 1e-12f);
  const size_t mo = (size_t)(blockIdx.x * 16 + r) * DIMC;
  #pragma unroll
  for (int j = 0; j < 16; ++j) {
    const int c = cg * 16 + j;
    float v = tile[r][c];
    float o = gamma[c] * (v * nx) + beta[c] + v * inv;
    o = o / (1.0f + __expf(-o));              // silu
    out[mo + c] = o;
  }
}

// ===========================================================================
// K2: xz = u_dir @ in_proj_w^T     (M=16384, N=512, K=128)
// u_dir[m,c] gathered from act_x with the reshape trick + per-direction flips:
//   x_flat[b,t,c] = act_flat[b, c*2048 + t];  dir&1 -> channel flip, dir&2 -> time flip
// one wave per 16x16 tile; grid sized exactly (no divergence before WMMA).
// ===========================================================================
__global__ void k_in_proj(const float* __restrict__ act,
                          const float* __restrict__ W,
                          float* __restrict__ xz)
{
  const int wave = (int)((blockIdx.x * (size_t)blockDim.x + threadIdx.x) >> 5);
  const int lane = threadIdx.x & 31;
  const int nt = wave & 31;            // 32 col tiles (N=512)
  const int mt = wave >> 5;            // 1024 row tiles
  const int half = lane >> 4, l15 = lane & 15;
  const int m0 = mt * 16, n0 = nt * 16;

  const int m   = m0 + l15;
  const int dir = m >> 12;
  const int b   = (m >> 11) & 1;
  const int t   = m & (SEQL - 1);
  const int te  = (dir & 2) ? (SEQL - 1 - t) : t;
  const bool cf = (dir & 1);
  const float* ab = act + (size_t)b * (DIMC * SEQL);
  const int n = n0 + l15;

  v8f acc = {};
  #pragma unroll 4
  for (int k0 = 0; k0 < DIMC; k0 += 4) {
    const int k  = k0 + half * 2;
    const int c0 = cf ? (DIMC - 1 - k) : k;
    const int c1 = cf ? (DIMC - 2 - k) : (k + 1);
    v2f a;  a.x = ab[(size_t)c0 * SEQL + te];  a.y = ab[(size_t)c1 * SEQL + te];
    v2f bb; bb.x = W[n * DIMC + k];            bb.y = W[n * DIMC + k + 1];
    acc = wmma4(acc, a, bb);
  }
  float* op = xz + (size_t)(m0 + (half << 3)) * 512 + n;
  #pragma unroll
  for (int r = 0; r < 8; ++r) op[(size_t)r * 512] = acc[r];
}

// ===========================================================================
// K3: causal depthwise conv-4 + bias + silu on xi_pre (= xz cols 0..255)
// ===========================================================================
__global__ void k_conv_silu(const float* __restrict__ xz,
                            const float* __restrict__ cw,
                            const float* __restrict__ cb,
                            float* __restrict__ xi)
{
  const size_t idx = blockIdx.x * (size_t)blockDim.x + threadIdx.x;
  if (idx >= (size_t)MROWS * DIN) return;
  const int d = (int)(idx & (DIN - 1));
  const size_t m = idx >> 8;
  const int t = (int)(m & (SEQL - 1));
  const size_t base = m - t;
  float s = cb[d];
  #pragma unroll
  for (int j = 0; j < 4; ++j) {
    const int tt = t - 3 + j;
    if (tt >= 0) s = fmaf(cw[d * 4 + j], xz[(base + tt) * 512 + d], s);
  }
  s = s / (1.0f + __expf(-s));
  xi[idx] = s;
}

// ===========================================================================
// K4: x_dbl = xi @ x_proj_w^T  (M=16384, N=40 padded to 48, K=256)
// cols 0..7 -> dt, 8..23 -> Bm, 24..39 -> Cm
// ===========================================================================
__global__ void k_x_proj(const float* __restrict__ xi,
                         const float* __restrict__ W,
                         float* __restrict__ dtb,
                         float* __restrict__ Bb,
                         float* __restrict__ Cb)
{
  const int wave = (int)((blockIdx.x * (size_t)blockDim.x + threadIdx.x) >> 5);
  const int lane = threadIdx.x & 31;
  const int nt = wave % 3;             // 3 col tiles (48)
  const int mt = wave / 3;             // 1024 row tiles
  const int half = lane >> 4, l15 = lane & 15;
  const int m0 = mt * 16;
  const int n  = nt * 16 + l15;
  const bool nvalid = (n < 40);

  const float* arow = xi + (size_t)(m0 + l15) * DIN;
  v8f acc = {};
  #pragma unroll 4
  for (int k0 = 0; k0 < DIN; k0 += 4) {
    const int k = k0 + half * 2;
    v2f a;  a.x = arow[k];  a.y = arow[k + 1];
    v2f bb;
    bb.x = nvalid ? W[n * DIN + k]     : 0.0f;
    bb.y = nvalid ? W[n * DIN + k + 1] : 0.0f;
    acc = wmma4(acc, a, bb);
  }
  #pragma unroll
  for (int r = 0; r < 8; ++r) {
    const size_t row = m0 + (half << 3) + r;
    const float v = acc[r];
    if (n < 8)       dtb[row * DTR + n]       = v;
    else if (n < 24) Bb [row * DST + (n - 8)]  = v;
    else if (n < 40) Cb [row * DST + (n - 24)] = v;
  }
}

// ===========================================================================
// K5a: delta = softplus(dt @ dt_proj_w^T + dt_proj_b)   (tiny K=8 -> scalar)
// ===========================================================================
__global__ void k_delta(const float* __restrict__ dtb,
                        const float* __restrict__ dtw,
                        const float* __restrict__ dtbias,
                        float* __restrict__ delta)
{
  const size_t idx = blockIdx.x * (size_t)blockDim.x + threadIdx.x;
  if (idx >= (size_t)MROWS * DIN) return;
  const int d = (int)(idx & (DIN - 1));
  const size_t m = idx >> 8;
  float s = dtbias[d];
  #pragma unroll
  for (int r = 0; r < DTR; ++r) s = fmaf(dtb[m * DTR + r], dtw[d * DTR + r], s);
  delta[idx] = (s > 20.0f) ? s : log1pf(__expf(s));
}

// ===========================================================================
// K5b: selective scan. One thread per (dir,b,d,s): scalar recurrence
//   h = exp(delta*A[d,s])*h + delta*B[s]*xi ;  y = sum_s h*C[s]
// 16-lane xor-shuffle reduction; s==0 lane applies +xi*D and *silu(z) and
// writes y_gated in place over the dead xi_pre half of xz (stride 512).
// block = 256 threads = (16 channels x 16 states); grid = 8 seqs * 16 groups.
// ===========================================================================
__global__ void k_scan(const float* __restrict__ delta,
                       const float* __restrict__ xi,
                       const float* __restrict__ Bb,
                       const float* __restrict__ Cb,
                       const float* __restrict__ A_log,
                       const float* __restrict__ Dvec,
                       float* __restrict__ xz)   // read z (cols 256..511), write y (cols 0..255)
{
  const int db = blockIdx.x >> 4;            // (dir*2+b) : 0..7
  const int cg = blockIdx.x & 15;            // channel group
  const int tid = threadIdx.x;
  const int d = cg * 16 + (tid >> 4);
  const int s = tid & 15;

  const float A  = -__expf(A_log[d * DST + s]);
  const float Dv = Dvec[d];
  const size_t mbase = (size_t)db * SEQL;
  float h = 0.0f;

  for (int t = 0; t < SEQL; ++t) {
    const size_t m = mbase + t;
    const float dl = delta[m * DIN + d];
    const float xv = xi[m * DIN + d];
    const float bv = Bb[m * DST + s];
    const float cv = Cb[m * DST + s];
    const float dA = __expf(dl * A);
    h = fmaf(dA, h, dl * bv * xv);
    float acc = h * cv;
    acc += __shfl_xor(acc, 8);
    acc += __shfl_xor(acc, 4);
    acc += __shfl_xor(acc, 2);
    acc += __shfl_xor(acc, 1);
    if (s == 0) {
      const float zv = xz[m * 512 + DIN + d];
      const float g  = zv / (1.0f + __expf(-zv));
      xz[m * 512 + d] = (acc + xv * Dv) * g;
    }
  }
}

// ===========================================================================
// K6: mamba_out = y_gated @ out_proj_w^T  (M=16384, N=128, K=256, lda=512)
// ===========================================================================
__global__ void k_out_proj(const float* __restrict__ xz,
                           const float* __restrict__ W,
                           float* __restrict__ mo)
{
  const int wave = (int)((blockIdx.x * (size_t)blockDim.x + threadIdx.x) >> 5);
  const int lane = threadIdx.x & 31;
  const int nt = wave & 7;             // 8 col tiles (N=128)
  const int mt = wave >> 3;            // 1024 row tiles
  const int half = lane >> 4, l15 = lane & 15;
  const int m0 = mt * 16, n0 = nt * 16;
  const int n = n0 + l15;

  const float* arow = xz + (size_t)(m0 + l15) * 512;
  v8f acc = {};
  #pragma unroll 4
  for (int k0 = 0; k0 < DIN; k0 += 4) {
    const int k = k0 + half * 2;
    v2f a;  a.x = arow[k];           a.y = arow[k + 1];
    v2f bb; bb.x = W[n * DIN + k];   bb.y = W[n * DIN + k + 1];
    acc = wmma4(acc, a, bb);
  }
  float* op = mo + (size_t)(m0 + (half << 3)) * DIMC + n;
  #pragma unroll
  for (int r = 0; r < 8; ++r) op[(size_t)r * DIMC] = acc[r];
}

// ===========================================================================
// K7: un-flip + sum 4 directions + *temp + swapaxes/reshape permutation
// gate[b, n*128+c] = temp * sum_dir mo[dir,b,t_d,c_d], with i=n*128+c,
// t_m = i%2048, c_m = i/2048.
// ===========================================================================
__global__ void k_combine(const float* __restrict__ mo,
                          const float* __restrict__ temp,
                          float* __restrict__ gate)
{
  const size_t idx = blockIdx.x * (size_t)blockDim.x + threadIdx.x;
  if (idx >= (size_t)NB * SEQL * DIMC) return;
  const int b = (int)(idx >> 18);
  const int i = (int)(idx & (SEQL * DIMC - 1));
  const int tm = i & (SEQL - 1);
  const int cm = i >> 11;
  float s = 0.0f;
  #pragma unroll
  for (int dir = 0; dir < NDIR; ++dir) {
    const int td = (dir & 2) ? (SEQL - 1 - tm) : tm;
    const int cd = (dir & 1) ? (DIMC - 1 - cm) : cm;
    s += mo[(((size_t)dir * NB + b) * SEQL + td) * DIMC + cd];
  }
  gate[idx] = s * temp[0];
}

// ===========================================================================
// K8a: per-(b,n) cosine similarity (one wave per row)
// ===========================================================================
__global__ void k_cos(const float* __restrict__ gate,
                      const float* __restrict__ act,
                      float* __restrict__ cosb)
{
  const int wave = (int)((blockIdx.x * (size_t)blockDim.x + threadIdx.x) >> 5);
  const int lane = threadIdx.x & 31;
  const int b = wave >> 11, n = wave & (SEQL - 1);
  const float* o = gate + ((size_t)b * SEQL + n) * DIMC;
  const float* a = act  + ((size_t)b * SEQL + n) * DIMC;
  float dot = 0.f, no = 0.f, na = 0.f;
  #pragma unroll
  for (int j = 0; j < 4; ++j) {
    const int c = lane * 4 + j;
    const float ov = o[c], av = a[c];
    dot = fmaf(ov, av, dot); no = fmaf(ov, ov, no); na = fmaf(av, av, na);
  }
  for (int off = 16; off; off >>= 1) {
    dot += __shfl_xor(dot, off); no += __shfl_xor(no, off); na += __shfl_xor(na, off);
  }
  if (lane == 0)
    cosb[b * SEQL + n] = dot / (fmaxf(sqrtf(no), 1e-8f) * fmaxf(sqrtf(na), 1e-8f));
}

// K8b: gate = weight_gate*sigmoid(mean_b cos)*gate + (1-w)*act
__global__ void k_gate(const float* __restrict__ cosb,
                       const float* __restrict__ wg,
                       const float* __restrict__ act,
                       float* __restrict__ gate)
{
  const size_t idx = blockIdx.x * (size_t)blockDim.x + threadIdx.x;
  if (idx >= (size_t)NB * SEQL * DIMC) return;
  const int n = (int)((idx & (SEQL * DIMC - 1)) >> 7);
  const float cm = 0.5f * (cosb[n] + cosb[SEQL + n]);
  const float w  = 1.0f / (1.0f + __expf(-cm));
  gate[idx] = wg[0] * w * gate[idx] + (1.0f - w) * act[idx];
}

// ===========================================================================
extern "C" void kernel_launch(void* const* d_in, const int* in_sizes, int n_in,
                              void* d_out, int out_size, void* d_ws, size_t ws_size,
                              hipStream_t stream)
{
  (void)in_sizes; (void)n_in; (void)out_size; (void)ws_size;
  const float* x        = (const float*)d_in[0];
  const float* nin_w    = (const float*)d_in[1];
  const float* nin_b    = (const float*)d_in[2];
  const float* nin2_w   = (const float*)d_in[3];
  const float* nin2_b   = (const float*)d_in[4];
  const float* g1_gamma = (const float*)d_in[5];
  const float* g1_beta  = (const float*)d_in[6];
  const float* g2_gamma = (const float*)d_in[7];
  const float* g2_beta  = (const float*)d_in[8];
  const float* temp     = (const float*)d_in[9];
  const float* wgate    = (const float*)d_in[10];
  const float* in_proj_w= (const float*)d_in[11];
  const float* conv_w   = (const float*)d_in[12];
  const float* conv_b   = (const float*)d_in[13];
  const float* x_proj_w = (const float*)d_in[14];
  const float* dt_proj_w= (const float*)d_in[15];
  const float* dt_proj_b= (const float*)d_in[16];
  const float* A_log    = (const float*)d_in[17];
  const float* ssm_D    = (const float*)d_in[18];
  const float* out_proj_w=(const float*)d_in[19];
  float* out = (float*)d_out;

  // workspace layout (floats) — everything stays resident in the 192MB L2
  float* f     = (float*)d_ws;
  float* act   = f;                         // 4096*128
  float* xz    = act   + (size_t)MACT*DIMC; // 16384*512 (xi_pre | z ; later y_gated | z)
  float* xi    = xz    + (size_t)MROWS*512; // 16384*256
  float* delta = xi    + (size_t)MROWS*DIN; // 16384*256
  float* dtb   = delta + (size_t)MROWS*DIN; // 16384*8
  float* Bb    = dtb   + (size_t)MROWS*DTR; // 16384*16
  float* Cb    = Bb    + (size_t)MROWS*DST; // 16384*16
  float* mo    = Cb    + (size_t)MROWS*DST; // 16384*128
  float* gate  = mo    + (size_t)MROWS*DIMC;// 4096*128
  float* cosb  = gate  + (size_t)MACT*DIMC; // 4096

  // 1) nin + GRN + silu -> act_x
  k_nin_grn_silu<<<MACT/16, 128, 0, stream>>>(x, 0, nin_w, nin_b, g1_gamma, g1_beta, act);
  // 2) in_proj for all 4 directions at once (flips folded into the gather)
  k_in_proj<<<(MROWS/16)*(512/16)/8, 256, 0, stream>>>(act, in_proj_w, xz);
  // 3) causal conv + silu
  k_conv_silu<<<(MROWS*DIN)/256, 256, 0, stream>>>(xz, conv_w, conv_b, xi);
  // 4) x_proj -> dt/B/C
  k_x_proj<<<(MROWS/16)*3/8, 256, 0, stream>>>(xi, x_proj_w, dtb, Bb, Cb);
  // 5a) delta = softplus(dt_proj)
  k_delta<<<(MROWS*DIN)/256, 256, 0, stream>>>(dtb, dt_proj_w, dt_proj_b, delta);
  // 5b) selective scan (writes y*silu(z) into xz cols 0..255)
  k_scan<<<8*16, 256, 0, stream>>>(delta, xi, Bb, Cb, A_log, ssm_D, xz);
  // 6) out_proj
  k_out_proj<<<(MROWS/16)*(DIMC/16)/8, 256, 0, stream>>>(xz, out_proj_w, mo);
  // 7) merge directions + temp + layout permutation
  k_combine<<<(NB*SEQL*DIMC)/256, 256, 0, stream>>>(mo, temp, gate);
  // 8) cosine gating
  k_cos<<<(MACT)/8, 256, 0, stream>>>(gate, act, cosb);
  k_gate<<<(NB*SEQL*DIMC)/256, 256, 0, stream>>>(cosb, wgate, act, gate);
  // 9) nin2 + GRN + silu -> output
  k_nin_grn_silu<<<MACT/16, 128, 0, stream>>>(gate, 1, nin2_w, nin2_b, g2_gamma, g2_beta, out);
}